// Mechanism_206158430747
// MI455X (gfx1250) — compile-verified
//
#include <hip/hip_runtime.h>

typedef __attribute__((ext_vector_type(2))) float v2f;
typedef __attribute__((ext_vector_type(8))) float v8f;

#define KMAIN 511
#define DDIM  64
#define BATCH 2048
#define KOUT  512   // KMAIN + null column

__device__ __forceinline__ float clip_softplus(float x) {
    // softplus, numerically stable, clipped to [0, 10]
    float sp = fmaxf(x, 0.0f) + log1pf(expf(-fabsf(x)));
    return fminf(fmaxf(sp, 0.0f), 10.0f);
}

// ---------------------------------------------------------------------------
// Kernel 1: per-k scaled weights c[k,d] = (softmax(logits[k])[d]+1e-10)*exp(clip(ld))
//           and beta[k]; block 511..  block 511 handles the null element.
// grid = 512 blocks, 64 threads (2 wave32s) each.
// ---------------------------------------------------------------------------
__global__ void precompute_kernel(const float* __restrict__ log_density,
                                  const float* __restrict__ logits,
                                  const float* __restrict__ beta_raw,
                                  const float* __restrict__ null_log_density,
                                  const float* __restrict__ null_logits,
                                  const float* __restrict__ null_beta_raw,
                                  float* __restrict__ cw,     // [KOUT*DDIM]; cw[KMAIN*DDIM] = c_null
                                  float* __restrict__ beta) { // [KOUT]
    __shared__ float red[DDIM];
    const int k = blockIdx.x;
    const int d = threadIdx.x;

    const float* lg = (k < KMAIN) ? (logits + k * DDIM) : null_logits;
    float x = lg[d];

    // max-reduce over D
    red[d] = x; __syncthreads();
    for (int s = 32; s > 0; s >>= 1) {
        if (d < s) red[d] = fmaxf(red[d], red[d + s]);
        __syncthreads();
    }
    float mx = red[0]; __syncthreads();

    float e = expf(x - mx);
    red[d] = e; __syncthreads();
    for (int s = 32; s > 0; s >>= 1) {
        if (d < s) red[d] += red[d + s];
        __syncthreads();
    }
    float sum = red[0];

    if (k < KMAIN) {
        float w  = e / sum;
        float ld = fminf(fmaxf(log_density[k * DDIM + d], -10.0f), 0.0f);
        // exp(M)*exp(log(w+1e-10)+ld - M) == (w+1e-10)*exp(ld)
        cw[k * DDIM + d] = (w + 1e-10f) * expf(ld);
        if (d == 0) beta[k] = clip_softplus(beta_raw[k]);
    } else if (d == 0) {
        float w0  = e / sum;  // thread 0 holds exp(x[0]-mx)
        float ld0 = fminf(fmaxf(null_log_density[0], -10.0f), 0.0f);
        cw[KMAIN * DDIM] = (w0 + 1e-10f) * expf(ld0);
        beta[KMAIN]      = clip_softplus(null_beta_raw[0]);
    }
}

// ---------------------------------------------------------------------------
// Kernel 2: U[b,k] = dot(vals[b,k,:], c[k,:]) - beta[k] via V_WMMA_F32_16X16X4_F32.
// One wave handles (16-b tile, one k) at a time: A = vals tile (16x4 per step),
// B = c[k] chunk broadcast into all 16 columns; column 0 of C = 16 dot products.
// grid = (B/16, 8) blocks of 256 threads (8 waves); wave w of block (bx,ky)
// covers k = w + 8*ky, stepping by 64.
// ---------------------------------------------------------------------------
__global__ void menu_util_kernel(const float* __restrict__ vals,
                                 const float* __restrict__ null_vals,
                                 const float* __restrict__ cw,
                                 const float* __restrict__ beta,
                                 float* __restrict__ out) {
    const int b0   = blockIdx.x * 16;
    const int lane = threadIdx.x & 31;
    const int wave = threadIdx.x >> 5;   // 0..7
    const int half = lane >> 4;          // ISA A/B layout: lanes 16-31 carry K+2
    const int m    = lane & 15;          // row (b) within the 16-wide tile

    // null column (k = 511), once per b-tile
    if (blockIdx.y == 0 && threadIdx.x < 16) {
        const float cn = cw[KMAIN * DDIM];
        const float bn = beta[KMAIN];
        const int b = b0 + threadIdx.x;
        out[(size_t)b * KOUT + KMAIN] = cn * null_vals[b] - bn;
    }

    const size_t row_stride = (size_t)KMAIN * DDIM;
    const float* arow0 = vals + (size_t)(b0 + m) * row_stride + 2 * half;

    for (int k = wave + 8 * blockIdx.y; k < KMAIN; k += 64) {
        const float* a = arow0 + (size_t)k * DDIM;   // this lane's slice of vals row
        const float* w = cw + k * DDIM + 2 * half;   // same per-lane index map as A

        // prefetch next k-row for this wave (global_prefetch_b8)
        __builtin_prefetch(a + 64 * DDIM, 0, 3);

        v8f acc0 = {};
        v8f acc1 = {};
#pragma unroll
        for (int c = 0; c < 16; c += 2) {
            // vals streamed once -> non-temporal (keep c[] resident in L2)
            v2f a0 = __builtin_nontemporal_load((const v2f*)a + 2 * c);
            v2f a1 = __builtin_nontemporal_load((const v2f*)a + 2 * (c + 1));
            v2f w0 = *((const v2f*)w + 2 * c);
            v2f w1 = *((const v2f*)w + 2 * (c + 1));
            acc0 = __builtin_amdgcn_wmma_f32_16x16x4_f32(
                false, a0, false, w0, (short)0, acc0, false, false);
            acc1 = __builtin_amdgcn_wmma_f32_16x16x4_f32(
                false, a1, false, w1, (short)0, acc1, false, false);
        }

        // C/D layout: lane 0 (col 0) has rows b0..b0+7 in v0..v7, lane 16 rows b0+8..b0+15
        if (m == 0) {
            const float bk = beta[k];
            const int brow = b0 + 8 * half;
#pragma unroll
            for (int i = 0; i < 8; ++i)
                out[(size_t)(brow + i) * KOUT + k] = acc0[i] + acc1[i] - bk;
        }
    }
}

extern "C" void kernel_launch(void* const* d_in, const int* in_sizes, int n_in,
                              void* d_out, int out_size, void* d_ws, size_t ws_size,
                              hipStream_t stream) {
    const float* vals             = (const float*)d_in[0];
    const float* log_density      = (const float*)d_in[1];
    const float* logits           = (const float*)d_in[2];
    const float* beta_raw         = (const float*)d_in[3];
    const float* null_vals        = (const float*)d_in[4];
    const float* null_log_density = (const float*)d_in[5];
    const float* null_logits      = (const float*)d_in[6];
    const float* null_beta_raw    = (const float*)d_in[7];
    float* out = (float*)d_out;

    float* cw   = (float*)d_ws;          // KOUT*DDIM floats (131072 B)
    float* beta = cw + KOUT * DDIM;      // KOUT floats

    precompute_kernel<<<dim3(KOUT), dim3(DDIM), 0, stream>>>(
        log_density, logits, beta_raw, null_log_density, null_logits,
        null_beta_raw, cw, beta);

    menu_util_kernel<<<dim3(BATCH / 16, 8), dim3(256), 0, stream>>>(
        vals, null_vals, cw, beta, out);
}